// VariationalQuantumLayer_58901181497724
// MI455X (gfx1250) — compile-verified
//
#include <hip/hip_runtime.h>
#include <math.h>

#define NQ      16
#define DIM     65536        // 2^16
#define NLAYERS 6
#define NBATCH  128

typedef __attribute__((ext_vector_type(2))) float    v2f;
typedef __attribute__((ext_vector_type(8))) float    v8f;
typedef __attribute__((ext_vector_type(4))) unsigned u32x4;
typedef __attribute__((ext_vector_type(8))) int      i32x8;
typedef __attribute__((ext_vector_type(4))) int      i32x4;

// ---------------------------------------------------------------------------
// complex helpers (float2 = {re, im})
// ---------------------------------------------------------------------------
__device__ __forceinline__ float2 cmad2(float2 u, float2 a, float2 v, float2 b) {
    float2 r;
    r.x = u.x * a.x - u.y * a.y + v.x * b.x - v.y * b.y;
    r.y = u.x * a.y + u.y * a.x + v.x * b.y + v.y * b.x;
    return r;
}

// PennyLane Rot = RZ(omega) RY(theta) RZ(phi)
__device__ __forceinline__ void rot_u(float phi, float theta, float omega,
                                      float2* U /* u00,u01,u10,u11 */) {
    float ch = cosf(0.5f * theta), sh = sinf(0.5f * theta);
    float a = 0.5f * (phi + omega), b = 0.5f * (phi - omega);
    float ca = cosf(a), sa = sinf(a);
    float cb = cosf(b), sb = sinf(b);
    U[0] = make_float2( ca * ch, -sa * ch);   // ep*c
    U[1] = make_float2(-cb * sh, -sb * sh);   // -conj(em)*s
    U[2] = make_float2( cb * sh, -sb * sh);   // em*s
    U[3] = make_float2( ca * ch,  sa * ch);   // conj(ep)*c
}

// ---------------------------------------------------------------------------
// TDM: issue a Tensor-Data-Mover load of a (tile_d0 x tile_d1) tile of 8-byte
// elements into LDS.  1D copy: tile_d1 == 0.  Must be executed by ONE wave
// (EXEC is ignored by TDM; every wave that executes it issues a full copy).
// D# layout per CDNA5 ISA ch.8: group0 = {flags, lds_addr, global_addr, type},
// group1 = {mask/data_size, tensor dims, tile dims, dim0 stride}.
// This toolchain's builtin is the 6-arg form: (g0, g1, g2, g3, g4, cpol).
// ---------------------------------------------------------------------------
__device__ __forceinline__ void tdm_load_tile_b8(const void* gptr, unsigned lds_off,
                                                 unsigned tensor_d0, unsigned tensor_d1,
                                                 unsigned tile_d0, unsigned tile_d1,
                                                 unsigned d0_stride /* elements */) {
    const unsigned long long ga = (unsigned long long)gptr;
    u32x4 g0;
    g0.x = 1u;                                        // count=1, user descriptor
    g0.y = lds_off;                                   // lds_addr (byte offset)
    g0.z = (unsigned)ga;                              // global_addr[31:0]
    g0.w = ((unsigned)(ga >> 32) & 0x01FFFFFFu)       // global_addr[56:32]
         | (2u << 30);                                // type = 2 ("image")
    i32x8 g1;
    g1[0] = (int)(3u << 16);                          // workgroup_mask=0, data_size=3 (8B)
    g1[1] = (int)((tensor_d0 & 0xFFFFu) << 16);       // tensor_dim0[15:0]
    g1[2] = (int)((tensor_d0 >> 16) | ((tensor_d1 & 0xFFFFu) << 16));
    g1[3] = (int)((tensor_d1 >> 16) | (tile_d0 << 16));
    g1[4] = (int)(tile_d1);                           // tile_dim1 (0 = 1D), tile_dim2=0
    g1[5] = (int)(d0_stride);                         // tensor_dim0_stride[31:0]
    g1[6] = 0;
    g1[7] = 0;
    const i32x4 z4 = {0, 0, 0, 0};                    // groups 2/3 unused (<=2D)
    const i32x8 z8 = {0, 0, 0, 0, 0, 0, 0, 0};
    __builtin_amdgcn_tensor_load_to_lds(g0, g1, z4, z4, z8, 0);
}

// ---------------------------------------------------------------------------
// Kernel 0: zero the 128x16 output (accumulated atomically later)
// ---------------------------------------------------------------------------
__global__ void k_zero_out(float* __restrict__ out, int n) {
    int i = blockIdx.x * 256 + threadIdx.x;
    if (i < n) out[i] = 0.0f;
}

// ---------------------------------------------------------------------------
// Kernel 1: per-batch L2 normalization, write complex state (imag = 0)
// ---------------------------------------------------------------------------
__global__ void k_norm(const float* __restrict__ x, float2* __restrict__ psi, int b0) {
    __shared__ float red[256];
    const float* xb = x + (size_t)(b0 + blockIdx.x) * DIM;
    float2* pb = psi + (size_t)blockIdx.x * DIM;
    const int tid = threadIdx.x;

    float s = 0.0f;
    for (int i = tid; i < DIM; i += 256) { float v = xb[i]; s += v * v; }
    red[tid] = s;
    __syncthreads();
    for (int off = 128; off > 0; off >>= 1) {
        if (tid < off) red[tid] += red[tid + off];
        __syncthreads();
    }
    const float inv = 1.0f / sqrtf(red[0]);
    for (int i = tid; i < DIM; i += 256)
        pb[i] = make_float2(xb[i] * inv, 0.0f);
}

// ---------------------------------------------------------------------------
// Kernel 2: gates on bit positions 0..7 (wires 15..8), strides 1..128.
// TDM loads a contiguous 2048-element (16 KB) tile into LDS; 8 butterfly
// sweeps; coalesced write-back in place.   grid = nb*32, block = 256.
// ---------------------------------------------------------------------------
__global__ void k_gates_low(float2* __restrict__ psi, const float* __restrict__ wts,
                            int layer) {
    __shared__ float2 tile[2048];                 // 16 KB
    __shared__ float2 U[8][4];
    const int bloc = blockIdx.x >> 5;
    const int t    = blockIdx.x & 31;
    float2* base = psi + (size_t)bloc * DIM + (size_t)t * 2048;
    const int tid = threadIdx.x;

    if (tid < 8) {                                // bit = tid, wire = 15 - tid
        const float* p = wts + ((layer * NQ) + (15 - tid)) * 3;
        rot_u(p[0], p[1], p[2], U[tid]);
    }
    if (tid < 32) {                               // wave 0 issues the DMA
        tdm_load_tile_b8(base, (unsigned)(uintptr_t)&tile[0],
                         /*tensor_d0=*/2048, 0, /*tile_d0=*/2048, 0, 0);
        __builtin_amdgcn_s_wait_tensorcnt(0);
    }
    __syncthreads();

    #pragma unroll
    for (int k = 0; k < 8; ++k) {
        const float2 u00 = U[k][0], u01 = U[k][1], u10 = U[k][2], u11 = U[k][3];
        #pragma unroll
        for (int pi = 0; pi < 4; ++pi) {
            const int p  = tid + pi * 256;        // 1024 disjoint pairs
            const int i0 = ((p >> k) << (k + 1)) | (p & ((1 << k) - 1));
            const int i1 = i0 | (1 << k);
            const float2 a = tile[i0], b = tile[i1];
            tile[i0] = cmad2(u00, a, u01, b);
            tile[i1] = cmad2(u10, a, u11, b);
        }
        __syncthreads();
    }
    for (int i = tid; i < 2048; i += 256) base[i] = tile[i];
}

// ---------------------------------------------------------------------------
// Kernel 3: gates on bit positions 8..15 (wires 7..0), strides 256..32768,
// then the fused CNOT-ring permutation (scatter via its GF(2)-linear inverse)
// into the other buffer.  TDM performs the 2D strided gather: a 16x256 tile
// of 8B elements (rows of 128B, row stride 2 KB) packed into LDS.
// grid = nb*16, block = 256.
// ---------------------------------------------------------------------------
__global__ void k_gates_high_perm(const float2* __restrict__ src,
                                  float2* __restrict__ dst,
                                  const float* __restrict__ wts,
                                  int layer, int r) {
    __shared__ float2 tile[4096];                 // [hi*16 + j], 32 KB
    __shared__ float2 U[8][4];
    const int bloc    = blockIdx.x >> 4;
    const int lo_base = (blockIdx.x & 15) * 16;
    const float2* sb = src + (size_t)bloc * DIM;
    float2*       db = dst + (size_t)bloc * DIM;
    const int tid = threadIdx.x;

    if (tid < 8) {                                // bit = 8 + tid, wire = 7 - tid
        const float* p = wts + ((layer * NQ) + (7 - tid)) * 3;
        rot_u(p[0], p[1], p[2], U[tid]);
    }
    if (tid < 32) {                               // 2D TDM gather: 16 elems x 256 rows
        tdm_load_tile_b8(sb + lo_base, (unsigned)(uintptr_t)&tile[0],
                         /*tensor_d0=*/256, /*tensor_d1=*/256,
                         /*tile_d0=*/16, /*tile_d1=*/256, /*d0_stride=*/256);
        __builtin_amdgcn_s_wait_tensorcnt(0);
    }
    __syncthreads();

    #pragma unroll
    for (int k = 0; k < 8; ++k) {
        const float2 u00 = U[k][0], u01 = U[k][1], u10 = U[k][2], u11 = U[k][3];
        #pragma unroll
        for (int pi = 0; pi < 8; ++pi) {
            const int p  = tid + pi * 256;        // 2048 disjoint pairs
            const int j  = p & 15;
            const int ph = p >> 4;
            const int h0 = ((ph >> k) << (k + 1)) | (ph & ((1 << k) - 1));
            const int h1 = h0 | (1 << k);
            const int i0 = h0 * 16 + j, i1 = h1 * 16 + j;
            const float2 a = tile[i0], b = tile[i1];
            tile[i0] = cmad2(u00, a, u01, b);
            tile[i1] = cmad2(u10, a, u11, b);
        }
        __syncthreads();
    }

    // Scatter through the inverse of the composed CNOT ring:
    //   gather G = P0∘P1∘...∘P15  =>  scatter: apply P0, then P1, ..., P15.
    //   P_w: if bit(15-w) set, flip bit(15-((w+r)%16)).
    for (int n = tid; n < 4096; n += 256) {
        const int hi = n >> 4, j = n & 15;
        unsigned idx = (unsigned)(hi * 256 + lo_base + j);
        #pragma unroll
        for (int w = 0; w < 16; ++w) {
            const unsigned cb = 15u - (unsigned)w;
            const unsigned tb = 15u - (unsigned)((w + r) & 15);
            if ((idx >> cb) & 1u) idx ^= (1u << tb);
        }
        db[idx] = tile[n];                        // scattered 8B writes, L2-resident
    }
}

// ---------------------------------------------------------------------------
// Kernel 4: out[b, q] = sum_i |psi[b, i]|^2 * (1 - 2*bit_{15-q}(i))
// as 16x16xK f32 GEMM per 16-batch group via V_WMMA_F32_16X16X4_F32.
// Branchless tail masking (unconditional clamped loads * float mask) so the
// K-loop has no EXEC divergence.
// ---------------------------------------------------------------------------
#define KSPLIT 32

__global__ void k_sign_reduce(const float2* __restrict__ psi, float* __restrict__ out,
                              int b0, int nb) {
    const int g    = blockIdx.x / KSPLIT;
    const int kb   = blockIdx.x % KSPLIT;
    const int klen = DIM / KSPLIT;                // 2048 K per block
    const int wave = threadIdx.x >> 5;            // 8 waves -> 256 K each
    const int lane = threadIdx.x & 31;

    const int kwave = klen / 8;                   // 256
    const int k0    = kb * klen + wave * kwave;
    const int mb    = lane & 15;                  // batch row within group
    const int khalf = (lane >> 4) << 1;           // lanes 16-31 own K = {2,3}
    const int bl    = g * 16 + mb;                // chunk-local batch
    const float msk = (bl < nb) ? 1.0f : 0.0f;    // branchless tail mask
    const float2* pb = psi + (size_t)(bl < nb ? bl : 0) * DIM;
    const int q    = lane & 15;                   // qubit column for B
    const int qs   = 15 - q;

    v8f c = {0.f, 0.f, 0.f, 0.f, 0.f, 0.f, 0.f, 0.f};

    #pragma unroll 4
    for (int ks = 0; ks < kwave; ks += 4) {
        const int kk = k0 + ks + khalf;
        const float2 p0 = pb[kk];
        const float2 p1 = pb[kk + 1];
        v2f a, b;
        a[0] = msk * (p0.x * p0.x + p0.y * p0.y);
        a[1] = msk * (p1.x * p1.x + p1.y * p1.y);
        b[0] = 1.0f - 2.0f * (float)(((unsigned)kk       >> qs) & 1u);
        b[1] = 1.0f - 2.0f * (float)(((unsigned)(kk + 1) >> qs) & 1u);
        c = __builtin_amdgcn_wmma_f32_16x16x4_f32(
                false, a, false, b, (short)0, c, false, false);
    }

    __shared__ float red[256];
    red[threadIdx.x] = 0.0f;
    __syncthreads();
    #pragma unroll
    for (int j = 0; j < 8; ++j) {
        const int m = j + ((lane >> 4) << 3);     // VGPR j: M=j (lanes 0-15), M=j+8
        const int n = lane & 15;
        atomicAdd(&red[m * 16 + n], c[j]);
    }
    __syncthreads();

    const int m = threadIdx.x >> 4, n = threadIdx.x & 15;
    if (g * 16 + m < nb)
        atomicAdd(&out[(size_t)(b0 + g * 16 + m) * 16 + n], red[threadIdx.x]);
}

// ---------------------------------------------------------------------------
// Host launcher.  Ping-pong state buffers in d_ws (1 MB per batch for both);
// batches chunked on ws_size.  All launches on `stream`; graph-capture safe.
// ---------------------------------------------------------------------------
extern "C" void kernel_launch(void* const* d_in, const int* in_sizes, int n_in,
                              void* d_out, int out_size, void* d_ws, size_t ws_size,
                              hipStream_t stream) {
    (void)in_sizes; (void)n_in; (void)out_size;
    const float* x   = (const float*)d_in[0];     // (128, 65536) f32
    const float* wts = (const float*)d_in[1];     // (6, 16, 3)   f32
    float* out = (float*)d_out;                   // (128, 16)    f32

    k_zero_out<<<(NBATCH * NQ + 255) / 256, 256, 0, stream>>>(out, NBATCH * NQ);

    const size_t per_batch = (size_t)DIM * sizeof(float2);        // 512 KB
    int nb_max = (int)(ws_size / (2 * per_batch));
    if (nb_max < 1) nb_max = 1;
    if (nb_max > NBATCH) nb_max = NBATCH;
    if (nb_max >= 16) nb_max &= ~15;              // keep full WMMA batch groups

    float2* buf0 = (float2*)d_ws;
    float2* buf1 = buf0 + (size_t)nb_max * DIM;

    for (int b0 = 0; b0 < NBATCH; b0 += nb_max) {
        const int nb = (NBATCH - b0 < nb_max) ? (NBATCH - b0) : nb_max;

        k_norm<<<nb, 256, 0, stream>>>(x, buf0, b0);

        float2* cur = buf0;
        float2* oth = buf1;
        for (int l = 0; l < NLAYERS; ++l) {
            const int r = l % (NQ - 1) + 1;
            k_gates_low<<<nb * 32, 256, 0, stream>>>(cur, wts, l);
            k_gates_high_perm<<<nb * 16, 256, 0, stream>>>(cur, oth, wts, l, r);
            float2* t = cur; cur = oth; oth = t;
        }

        const int ngroups = (nb + 15) / 16;
        k_sign_reduce<<<ngroups * KSPLIT, 256, 0, stream>>>(cur, out, b0, nb);
    }
}